// DataReloadingQNN_69578470195741
// MI455X (gfx1250) — compile-verified
//
#include <hip/hip_runtime.h>

typedef float v2f __attribute__((ext_vector_type(2)));
typedef float v8f __attribute__((ext_vector_type(8)));

#define N_QUBITS 11
#define DIM 2048          // 2^11
#define N_LAYERS 6
#define BATCH 8192
#define WAVES_PER_BLOCK 2
#define THREADS (WAVES_PER_BLOCK * 32)

// ---------------------------------------------------------------------------
// CNOT-chain folding. The per-layer CNOT chain (c -> c+1, c = 0..9) is linear
// over GF(2): it maps bit p of the basis index to XOR of bits p..10, i.e.
// f = I+S+...+S^10 (S = shift toward LSB). Its inverse is the Gray-code map
// g(j) = j ^ (j>>1). After l layers the logical amplitude i lives at physical
// LDS slot tau_l(i) = g^l(i) = prod_{bit m of l} (I + S^(2^m)) applied to i.
// tau is XOR-linear, so tau(row|col) = tau(row) ^ tau(col) for disjoint fields.
// ---------------------------------------------------------------------------
__device__ __forceinline__ int perm_of(int i, int m1, int m2, int m4) {
    i ^= (i >> 1) & m1;    // masks are 0 or -1 (uniform per layer)
    i ^= (i >> 2) & m2;
    i ^= (i >> 4) & m4;
    return i;
}

// ---------------------------------------------------------------------------
// Logical address fields (in float2 units) of state element (row b, col c).
// G0: qubits 0-3 (bits 10..7):   i = b*128 + c
// G1: qubits 4-7 (bits 6..3):    i = x*128 + b*8 + y,  c = x*8 + y
// G2: qubits 7-10 (bits 3..0):   i = c*16 + b   (identity in qubit-7 slot)
// Row/col fields occupy disjoint bit ranges in every group.
// ---------------------------------------------------------------------------
template <int G>
__device__ __forceinline__ int row_field(int r) {
    if constexpr (G == 0) return r << 7;
    else if constexpr (G == 1) return r << 3;
    else return r;
}
template <int G>
__device__ __forceinline__ int col_field(int c) {
    if constexpr (G == 0) return c;
    else if constexpr (G == 1) return ((c >> 3) << 7) | (c & 7);
    else return c << 4;
}

// Wave-local LDS ordering point: each wave owns its sample's LDS region and
// the LDS pipe is in-order per wave, so no s_barrier is needed.
__device__ __forceinline__ void lds_sync() {
    __builtin_amdgcn_wave_barrier();
    __builtin_amdgcn_fence(__ATOMIC_SEQ_CST, "workgroup");
    __builtin_amdgcn_wave_barrier();
}

// ---------------------------------------------------------------------------
// Kernel 1: build the 18 shared 16x16 group unitaries (6 layers x 3 groups).
// gate_ws layout: float2 gate[(l*3+g)*256 + row*16 + col] = (re, im)
// ---------------------------------------------------------------------------
__global__ void build_gates_kernel(const float* __restrict__ params,
                                   float2* __restrict__ gate_ws) {
    int t = threadIdx.x;
    if (t >= N_LAYERS * 3) return;
    int l = t / 3, g = t % 3;

    float ur[4][2][2], ui[4][2][2];
    #pragma unroll
    for (int j = 0; j < 4; ++j) {
        bool ident = (g == 2 && j == 0);
        int q = (g == 2) ? (7 + j) : (g * 4 + j);
        if (ident) {
            ur[j][0][0] = 1.f; ur[j][0][1] = 0.f; ur[j][1][0] = 0.f; ur[j][1][1] = 1.f;
            ui[j][0][0] = 0.f; ui[j][0][1] = 0.f; ui[j][1][0] = 0.f; ui[j][1][1] = 0.f;
        } else {
            float p0 = params[(l * N_QUBITS + q) * 3 + 0];
            float p1 = params[(l * N_QUBITS + q) * 3 + 1];
            float p2 = params[(l * N_QUBITS + q) * 3 + 2];
            // a = e^{-i p2/2}, b = e^{-i p0/2}, RY(p1) = [[c,-s],[s,c]]
            float ar = __cosf(0.5f * p2), ai = -__sinf(0.5f * p2);
            float br = __cosf(0.5f * p0), bi = -__sinf(0.5f * p0);
            float c  = __cosf(0.5f * p1), s  =  __sinf(0.5f * p1);
            float abr  = ar * br - ai * bi;   // a*b
            float abi  = ar * bi + ai * br;
            float acbr = ar * br + ai * bi;   // a*conj(b)
            float acbi = ai * br - ar * bi;
            // U = RZ(p2)*RY(p1)*RZ(p0)
            ur[j][0][0] =  c * abr;   ui[j][0][0] =  c * abi;
            ur[j][0][1] = -s * acbr;  ui[j][0][1] = -s * acbi;
            ur[j][1][0] =  s * acbr;  ui[j][1][0] = -s * acbi;
            ur[j][1][1] =  c * abr;   ui[j][1][1] = -c * abi;
        }
    }

    float2* dst = gate_ws + (size_t)t * 256;
    for (int idx = 0; idx < 256; ++idx) {
        int row = idx >> 4, col = idx & 15;
        float re = 1.f, im = 0.f;
        #pragma unroll
        for (int j = 0; j < 4; ++j) {
            int rb = (row >> (3 - j)) & 1;
            int cb = (col >> (3 - j)) & 1;
            float xr = ur[j][rb][cb], xi = ui[j][rb][cb];
            float nr = re * xr - im * xi;
            float ni = re * xi + im * xr;
            re = nr; im = ni;
        }
        dst[idx] = make_float2(re, im);
    }
}

// ---------------------------------------------------------------------------
// Load A-fragments of a complex 16x16 gate: lane holds G[M=lo][K=4kc+2hi+v].
// ---------------------------------------------------------------------------
__device__ __forceinline__ void load_gate_frags(const float2* __restrict__ G,
                                                int hi, int lo,
                                                v2f aRe[4], v2f aIm[4], v2f aImN[4]) {
    #pragma unroll
    for (int kc = 0; kc < 4; ++kc) {
        #pragma unroll
        for (int v = 0; v < 2; ++v) {
            int k = 4 * kc + 2 * hi + v;
            float2 gv = G[lo * 16 + k];
            aRe[kc][v]  = gv.x;
            aIm[kc][v]  = gv.y;
            aImN[kc][v] = -gv.y;
        }
    }
}

// ---------------------------------------------------------------------------
// Apply a 16x16 gate (complex if CPLX, else real) to the state via WMMA,
// addressing LDS through the accumulated CNOT permutation tau (m1,m2,m4).
// Tile body is staged: addresses -> batched ds_loads -> repack -> WMMA -> store
// so the LDS pipe can run ahead with partial dscnt waits.
// ---------------------------------------------------------------------------
template <int G, bool CPLX>
__device__ __forceinline__ void apply_group(float2* st, int hi, int lo,
                                            int m1, int m2, int m4,
                                            const v2f aRe[4], const v2f aIm[4],
                                            const v2f aImN[4]) {
    // Permuted row fields (constant across tiles).
    int rowLd[8];
    #pragma unroll
    for (int k8 = 0; k8 < 8; ++k8) {      // k8 = kc*2 + v
        int krow = 4 * (k8 >> 1) + 2 * hi + (k8 & 1);
        rowLd[k8] = perm_of(row_field<G>(krow), m1, m2, m4);
    }
    int rowSt[8];
    #pragma unroll
    for (int r = 0; r < 8; ++r)
        rowSt[r] = perm_of(row_field<G>(r + 8 * hi), m1, m2, m4);

    #pragma unroll 2
    for (int t = 0; t < 8; ++t) {
        const int col  = 16 * t + lo;                           // B/D column N
        const int pcol = perm_of(col_field<G>(col), m1, m2, m4);

        // Phase 1: all addresses.
        int adr[8];
        #pragma unroll
        for (int k8 = 0; k8 < 8; ++k8) adr[k8] = pcol ^ rowLd[k8];

        // Phase 2: batched loads.
        float2 sv[8];
        #pragma unroll
        for (int k8 = 0; k8 < 8; ++k8) sv[k8] = st[adr[k8]];    // ds_load_b64

        // Phase 3: repack into WMMA B operands.
        v2f bSr[4], bSi[4];
        #pragma unroll
        for (int kc = 0; kc < 4; ++kc) {
            bSr[kc][0] = sv[kc * 2 + 0].x;  bSi[kc][0] = sv[kc * 2 + 0].y;
            bSr[kc][1] = sv[kc * 2 + 1].x;  bSi[kc][1] = sv[kc * 2 + 1].y;
        }

        v8f Dr = {}; v8f Di = {};
        #pragma unroll
        for (int kc = 0; kc < 4; ++kc) {
            // Dr = Gr*Sr + (-Gi)*Si ;  Di = Gr*Si + Gi*Sr
            Dr = __builtin_amdgcn_wmma_f32_16x16x4_f32(false, aRe[kc], false, bSr[kc], (short)0, Dr, false, false);
            Di = __builtin_amdgcn_wmma_f32_16x16x4_f32(false, aRe[kc], false, bSi[kc], (short)0, Di, false, false);
            if constexpr (CPLX) {
                Dr = __builtin_amdgcn_wmma_f32_16x16x4_f32(false, aImN[kc], false, bSi[kc], (short)0, Dr, false, false);
                Di = __builtin_amdgcn_wmma_f32_16x16x4_f32(false, aIm[kc],  false, bSr[kc], (short)0, Di, false, false);
            }
        }

        #pragma unroll
        for (int r = 0; r < 8; ++r)
            st[pcol ^ rowSt[r]] = make_float2(Dr[r], Di[r]);    // ds_store_b64
    }
    lds_sync();
}

// Build A-fragments of the per-sample real RY tensor product for group g.
__device__ __forceinline__ void build_ry_frags(const float* __restrict__ X, int samp,
                                               int g, int hi, int lo, v2f aR[4]) {
    float rc[4], rs[4];
    #pragma unroll
    for (int j = 0; j < 4; ++j) {
        bool ident = (g == 2 && j == 0);
        int q = (g == 2) ? (7 + j) : (g * 4 + j);
        float x = ident ? 0.f : X[samp * N_QUBITS + q];
        rc[j] = __cosf(0.5f * x);
        rs[j] = __sinf(0.5f * x);
    }
    #pragma unroll
    for (int kc = 0; kc < 4; ++kc) {
        #pragma unroll
        for (int v = 0; v < 2; ++v) {
            int k = 4 * kc + 2 * hi + v;
            float val = 1.f;
            #pragma unroll
            for (int j = 0; j < 4; ++j) {
                int rb = (lo >> (3 - j)) & 1;
                int cb = (k  >> (3 - j)) & 1;
                val *= (rb == cb) ? rc[j] : ((rb == 1) ? rs[j] : -rs[j]);
            }
            aR[kc][v] = val;
        }
    }
}

// ---------------------------------------------------------------------------
// Kernel 2: one wave per sample; state resident in LDS as float2; CNOTs are
// never materialized — they live entirely in the address permutation.
// ---------------------------------------------------------------------------
__launch_bounds__(THREADS)
__global__ void qnn_forward_kernel(const float* __restrict__ X,
                                   const float2* __restrict__ gate_ws,
                                   float2* __restrict__ out) {
    __shared__ float2 s_st[WAVES_PER_BLOCK][DIM];   // 32 KB

    const int w    = threadIdx.x >> 5;
    const int lane = threadIdx.x & 31;
    const int samp = blockIdx.x * WAVES_PER_BLOCK + w;
    float2* st = s_st[w];

    // |0...0>   (tau_0 = identity)
    for (int i = lane; i < DIM; i += 32) st[i] = make_float2(0.f, 0.f);
    if (lane == 0) st[0] = make_float2(1.f, 0.f);
    lds_sync();

    const int hi = lane >> 4;   // K/M half-select
    const int lo = lane & 15;   // A row M | B,D col N

    v2f aRe[4], aIm[4], aImN[4];

    for (int l = 0; l < N_LAYERS; ++l) {
        // tau_l = g^l: select shift terms from the bits of l (uniform masks)
        const int m1 = (l & 1) ? -1 : 0;
        const int m2 = (l & 2) ? -1 : 0;
        const int m4 = (l & 4) ? -1 : 0;

        const float2* Gl = gate_ws + (size_t)(l * 3) * 256;
        if (l + 1 < N_LAYERS)
            __builtin_prefetch(gate_ws + (size_t)((l + 1) * 3) * 256, 0, 3);

        load_gate_frags(Gl + 0 * 256, hi, lo, aRe, aIm, aImN);
        apply_group<0, true>(st, hi, lo, m1, m2, m4, aRe, aIm, aImN);
        load_gate_frags(Gl + 1 * 256, hi, lo, aRe, aIm, aImN);
        apply_group<1, true>(st, hi, lo, m1, m2, m4, aRe, aIm, aImN);
        load_gate_frags(Gl + 2 * 256, hi, lo, aRe, aIm, aImN);
        apply_group<2, true>(st, hi, lo, m1, m2, m4, aRe, aIm, aImN);

        if (l == N_LAYERS - 1) {
            // per-sample data re-upload: real RY tensor-product groups (tau_5)
            v2f aR[4];
            build_ry_frags(X, samp, 0, hi, lo, aR);
            apply_group<0, false>(st, hi, lo, m1, m2, m4, aR, aR, aR);
            build_ry_frags(X, samp, 1, hi, lo, aR);
            apply_group<1, false>(st, hi, lo, m1, m2, m4, aR, aR, aR);
            build_ry_frags(X, samp, 2, hi, lo, aR);
            apply_group<2, false>(st, hi, lo, m1, m2, m4, aR, aR, aR);
        }
        // CNOT chain: folded into tau — no data movement.
    }

    // Final state is at tau_6 = (I+S^2)(I+S^4): gather and write (B, 2048, 2).
    for (int i = lane; i < DIM; i += 32) {
        int p = perm_of(i, 0, -1, -1);
        out[(size_t)samp * DIM + i] = st[p];
    }
}

// ---------------------------------------------------------------------------
extern "C" void kernel_launch(void* const* d_in, const int* in_sizes, int n_in,
                              void* d_out, int out_size, void* d_ws, size_t ws_size,
                              hipStream_t stream) {
    const float* X      = (const float*)d_in[0];   // (8192, 11) f32
    const float* params = (const float*)d_in[1];   // (6, 11, 3) f32
    float2* out         = (float2*)d_out;          // (8192, 2048, 2) f32
    float2* gate_ws     = (float2*)d_ws;           // 18*256 float2 = 36 KB

    build_gates_kernel<<<1, 32, 0, stream>>>(params, gate_ws);
    qnn_forward_kernel<<<BATCH / WAVES_PER_BLOCK, THREADS, 0, stream>>>(X, gate_ws, out);
}